// DTWLoss_41377714929792
// MI455X (gfx1250) — compile-verified
//
#include <hip/hip_runtime.h>
#include <stdint.h>
#include <math.h>

// ---------------------------------------------------------------------------
// DTW loss on MI455X (gfx1250).
// One workgroup (1024 threads = 32 waves) per batch sample.
//  - Stage pred/targ basis columns (== loss columns) into LDS via the
//    Tensor Data Mover (tensor_load_to_lds, 2D descriptor, tile 2x1024 out of
//    a [1024 x 4] f32 tensor).
//  - Anti-diagonal wavefront DP, thread i owns row i. Three rotating diagonal
//    buffers in LDS with a +1 guard slot holding +inf (branch-free boundary).
//  - Per-cell backtrack direction (argmin index, jnp.argmin first-min
//    tie-break, inf at borders) packed 2 bits/cell into a 256 KB LDS matrix.
//  - Thread 0 backtracks entirely out of LDS and writes the per-batch loss.
//  - Tiny second kernel sums partials deterministically.
// ---------------------------------------------------------------------------

#define DTW_N 1024
#define DTW_M 1024
#define DTW_L (DTW_N + DTW_M - 1)          // 2047
#define DIR_WPR (DTW_M / 16)               // 64 u32 words per row (2 bits/cell)

// LDS layout (single byte array, offsets are also used as TDM lds_addr).
#define PB_OFF   0u                               // float2[1024]  : 8 KB
#define TB_OFF   8192u                            // float2[1024]  : 8 KB
#define DIAG_OFF 16384u                           // float[3][1025]: 12300 B
#define DIR_OFF  (DIAG_OFF + 3u * (DTW_N + 1) * 4u)   // 28684
#define SM_TOTAL (DIR_OFF + (uint32_t)DTW_N * DIR_WPR * 4u)  // 290828 B

typedef unsigned int u32x4 __attribute__((ext_vector_type(4)));
typedef int          i32x8 __attribute__((ext_vector_type(8)));
typedef int          i32x4 __attribute__((ext_vector_type(4)));

// Issue one TDM load: tile [2 x rows] (first 2 f32 of each 4-f32 row) from a
// row-major [rows x 4] f32 tensor at gaddr -> packed float2[rows] at lds_off.
__device__ __forceinline__ void tdm_load_basis(uint64_t gaddr, uint32_t lds_off,
                                               uint32_t rows) {
  u32x4 g0;
  g0.x = 1u;                                    // count=1, user mode, no gather
  g0.y = lds_off;                               // lds_addr (bytes)
  g0.z = (uint32_t)(gaddr & 0xffffffffu);       // global_addr[31:0]
  g0.w = (uint32_t)((gaddr >> 32) & 0x1ffffffu) // global_addr[56:32]
         | (2u << 30);                          // type = 2 ("image")
  i32x8 g1;
  g1[0] = (int)(2u << 16);          // workgroup_mask=0, data_size=2 (4 bytes)
  g1[1] = (int)(4u << 16);          // tensor_dim0 = 4 elements (bits 63:48)
  g1[2] = (int)(rows << 16);        // tensor_dim1 = rows       (bits 95:80)
  g1[3] = (int)(2u << 16);          // tile_dim0 = 2            (bits 127:112)
  g1[4] = (int)rows;                // tile_dim1 = rows; tile_dim2 = 0
  g1[5] = 4;                        // tensor_dim0_stride = 4 elements
  g1[6] = 0;
  g1[7] = 0;
  i32x4 z4 = {0, 0, 0, 0};
  i32x8 z8 = {0, 0, 0, 0, 0, 0, 0, 0};
  // clang-23 / therock-10.0 form: 6 args (g0, g1, g2, g3, extra, cpol).
  __builtin_amdgcn_tensor_load_to_lds(g0, g1, z4, z4, z8, 0);
}

__global__ __launch_bounds__(1024) void dtw_batch_kernel(
    const float* __restrict__ preds,   // [B, 1024, 4]
    const float* __restrict__ targs,   // [B, 1024, 4]
    const float* __restrict__ subcoef, // [2]
    float* __restrict__ partial)       // [B]
{
  __shared__ __align__(16) unsigned char smem[SM_TOTAL];

  const int tid = threadIdx.x;   // row index i
  const int b   = blockIdx.x;

  float2*   pB   = (float2*)(smem + PB_OFF);
  float2*   tB   = (float2*)(smem + TB_OFF);
  float*    dgb  = (float*)(smem + DIAG_OFF);
  uint32_t* dirM = (uint32_t*)(smem + DIR_OFF);

  // --- Stage basis/loss columns into LDS via the Tensor Data Mover ---------
  if (tid == 0) {
    tdm_load_basis((uint64_t)(uintptr_t)(preds + (size_t)b * DTW_N * 4),
                   PB_OFF, DTW_N);
    tdm_load_basis((uint64_t)(uintptr_t)(targs + (size_t)b * DTW_M * 4),
                   TB_OFF, DTW_M);
  }

  // Init rotating diagonal buffers (guard slot [0] stays +inf forever).
  for (int x = tid; x < 3 * (DTW_N + 1); x += 1024) dgb[x] = INFINITY;

  __builtin_amdgcn_s_wait_tensorcnt(0);  // non-issuing waves pass instantly
  __syncthreads();

  const float2 p = pB[tid];
  float* cur   = dgb;                      // diagonal k   (k=0 -> buf 0)
  float* prev2 = dgb + (DTW_N + 1);        // diagonal k-2
  float* prev1 = dgb + 2 * (DTW_N + 1);    // diagonal k-1
  uint32_t* dirRow = dirM + tid * DIR_WPR;
  uint32_t acc = 0;

  // --- Anti-diagonal wavefront: one barrier per diagonal -------------------
  for (int k = 0; k < DTW_L; ++k) {
    const int j = k - tid;
    if ((unsigned)j < (unsigned)DTW_M) {
      const float2 t = tB[j];
      const float dx = p.x - t.x;
      const float dy = p.y - t.y;
      const float c  = sqrtf(dx * dx + dy * dy);
      const float up = prev1[tid];       // D[i-1, j]
      const float dg = prev2[tid];       // D[i-1, j-1]
      const float lf = prev1[tid + 1];   // D[i,   j-1]
      // jnp.argmin([dg, up, lf]) with first-min tie-break (inf at borders).
      const uint32_t m = (dg <= up && dg <= lf) ? 0u : ((up <= lf) ? 1u : 2u);
      float best = fminf(fminf(up, dg), lf);
      best = (best == INFINITY) ? 0.0f : best;   // (0,0) seed
      cur[tid + 1] = c + best;
      acc |= m << ((j & 15) * 2);
      if ((j & 15) == 15) { dirRow[j >> 4] = acc; acc = 0; }
    } else {
      cur[tid + 1] = INFINITY;
    }
    __syncthreads();
    float* tmp = prev2; prev2 = prev1; prev1 = cur; cur = tmp;
  }

  // --- Backtrack fully inside LDS ------------------------------------------
  if (tid == 0) {
    const float s0 = subcoef[0];
    const float s1 = subcoef[1];
    int i = DTW_N - 1, j = DTW_M - 1;
    float loss = 0.0f;
    while (true) {
      const float2 pp = pB[i];
      const float2 tt = tB[j];
      loss += fabsf(pp.x - tt.x) * s0 + fabsf(pp.y - tt.y) * s1;
      if ((i | j) == 0) break;
      const uint32_t w = dirM[i * DIR_WPR + (j >> 4)];
      const uint32_t m = (w >> ((j & 15) * 2)) & 3u;
      i -= (m != 2u);   // diag or up
      j -= (m != 1u);   // diag or left
    }
    partial[b] = loss;
  }
}

__global__ void dtw_sum_kernel(const float* __restrict__ partial,
                               float* __restrict__ out, int B) {
  if (threadIdx.x == 0 && blockIdx.x == 0) {
    float s = 0.0f;
    for (int b = 0; b < B; ++b) s += partial[b];  // fixed order: deterministic
    out[0] = s;
  }
}

extern "C" void kernel_launch(void* const* d_in, const int* in_sizes, int n_in,
                              void* d_out, int out_size, void* d_ws, size_t ws_size,
                              hipStream_t stream) {
  const float* preds   = (const float*)d_in[0];
  const float* targs   = (const float*)d_in[1];
  const float* subcoef = (const float*)d_in[2];
  float* out     = (float*)d_out;
  float* partial = (float*)d_ws;            // B floats of scratch

  const int B = in_sizes[0] / (DTW_N * 4);  // preds is [B, N, 4]

  dtw_batch_kernel<<<B, 1024, 0, stream>>>(preds, targs, subcoef, partial);
  dtw_sum_kernel<<<1, 32, 0, stream>>>(partial, out, B);
}